// SVDNet_21174188769399
// MI455X (gfx1250) — compile-verified
//
#include <hip/hip_runtime.h>
#include <math.h>

typedef __attribute__((ext_vector_type(2))) float v2f;
typedef __attribute__((ext_vector_type(8))) float v8f;
typedef __attribute__((ext_vector_type(4))) unsigned int u32x4;
typedef __attribute__((ext_vector_type(8))) unsigned int u32x8;

#define EPS_BN 1e-5f
#define ORTHO_OUTER 10
#define ORTHO_NS 16

static __device__ inline v8f wmma4(v2f a, v2f b, v8f c) {
  // D = A(16x4) * B(4x16) + C, full f32
  return __builtin_amdgcn_wmma_f32_16x16x4_f32(false, a, false, b, (short)0, c,
                                               false, false);
}

static __device__ inline float dgelu(float x) {
  return 0.5f * x * (1.0f + erff(x * 0.70710678118654752f));
}
static __device__ inline float dsoftplus(float x) {
  return x > 0.f ? x + log1pf(expf(-x)) : log1pf(expf(x));
}

// ---------------- implicit-GEMM conv with WMMA f32 16x16x4 ----------------
// One wave computes a 16(Cout) x 16(pixel) tile. act: 0=bias, 1=bias+BN, 2=bias+BN+GELU
__global__ __launch_bounds__(256) void conv_wmma_kernel(
    const float* __restrict__ in, const float* __restrict__ wgt,
    const float* __restrict__ bias, const float* __restrict__ bng,
    const float* __restrict__ bnb, const float* __restrict__ bnm,
    const float* __restrict__ bnv, float* __restrict__ out, int Cin, int Cout,
    int H, int W, int ks, int act) {
  const int HW = H * W;
  const int ptiles = HW >> 4;
  const int ntiles = ptiles * (Cout >> 4);
  int tile = blockIdx.x * 8 + (threadIdx.x >> 5);
  if (tile >= ntiles) return;  // wave-uniform
  int lane = threadIdx.x & 31;
  int m = lane & 15, kh = lane >> 4;
  int pix0 = (tile % ptiles) << 4;
  int cout0 = (tile / ptiles) << 4;
  int p = pix0 + m;  // B-fragment column pixel (n = m)
  int ph = p / W, pw = p - ph * W;
  int kk2 = ks * ks, K = Cin * kk2, pad = ks >> 1;
  const float* wrow = wgt + (size_t)(cout0 + m) * K;  // A row (M = m)
  v8f acc = {};
  for (int k0 = 0; k0 < K; k0 += 4) {
    v2f a, b;
    for (int j = 0; j < 2; ++j) {
      int k = k0 + 2 * kh + j;  // A: lane half holds K pair {2*kh, 2*kh+1}
      float av = 0.f, bv = 0.f;
      if (k < K) {
        av = wrow[k];
        int cin = k / kk2, rem = k - cin * kk2;
        int hh = ph + rem / ks - pad;
        int ww = pw + rem % ks - pad;
        if (hh >= 0 && hh < H && ww >= 0 && ww < W)
          bv = in[(size_t)cin * HW + hh * W + ww];
      }
      a[j] = av;
      b[j] = bv;
    }
    acc = wmma4(a, b, acc);
  }
  for (int r = 0; r < 8; ++r) {  // C/D: lane half kh holds rows 8*kh..8*kh+7
    int c = cout0 + kh * 8 + r;
    float v = acc[r] + bias[c];
    if (act >= 1) v = (v - bnm[c]) * (bng[c] * rsqrtf(bnv[c] + EPS_BN)) + bnb[c];
    if (act == 2) v = dgelu(v);
    out[(size_t)c * HW + pix0 + m] = v;
  }
}

// ---------------- bandwidth-bound GEMV (one row per wave32) ----------------
// act: 0 none, 1 gelu, 2 relu, 3 softplus
__global__ __launch_bounds__(256) void gemv_kernel(
    const float* __restrict__ Wm, const float* __restrict__ bias,
    const float* __restrict__ x, float* __restrict__ y, int In, int Out,
    int act) {
  int wv = threadIdx.x >> 5, lane = threadIdx.x & 31;
  int row = blockIdx.x * 8 + wv;
  if (row >= Out) return;
  const float4* wr = (const float4*)(Wm + (size_t)row * In);
  const float4* xv = (const float4*)x;
  int n4 = In >> 2;
  float sum = 0.f;
  for (int i = lane; i < n4; i += 32) {
    __builtin_prefetch(wr + i + 64, 0, 1);
    float4 a = wr[i];
    float4 b = xv[i];
    sum += a.x * b.x + a.y * b.y + a.z * b.z + a.w * b.w;
  }
  for (int off = 16; off; off >>= 1) sum += __shfl_xor(sum, off, 32);
  if (lane == 0) {
    float v = sum + bias[row];
    if (act == 1) v = dgelu(v);
    else if (act == 2) v = fmaxf(v, 0.f);
    else if (act == 3) v = dsoftplus(v);
    y[row] = v;
  }
}

// ---------------- misc small kernels ----------------
__global__ void nhwc_to_chw_kernel(const float* __restrict__ x,
                                   float* __restrict__ out, int H, int W) {
  int i = blockIdx.x * 256 + threadIdx.x;
  int total = H * W;
  if (i >= total) return;
  out[i] = x[2 * i];
  out[total + i] = x[2 * i + 1];
}

__global__ void channel_mean_kernel(const float* __restrict__ in,
                                    float* __restrict__ out, int HW) {
  int c = blockIdx.x;
  __shared__ float red[256];
  float s = 0.f;
  for (int i = threadIdx.x; i < HW; i += 256) s += in[(size_t)c * HW + i];
  red[threadIdx.x] = s;
  __syncthreads();
  for (int o = 128; o; o >>= 1) {
    if (threadIdx.x < (unsigned)o) red[threadIdx.x] += red[threadIdx.x + o];
    __syncthreads();
  }
  if (threadIdx.x == 0) out[c] = red[0] / (float)HW;
}

__global__ void se_fc_kernel(const float* __restrict__ pooled,
                             const float* __restrict__ w1,
                             const float* __restrict__ b1,
                             const float* __restrict__ w2,
                             const float* __restrict__ b2,
                             float* __restrict__ att, int C, int C4) {
  __shared__ float hid[64];
  int t = threadIdx.x;
  if (t < C4) {
    float s = b1[t];
    for (int j = 0; j < C; ++j) s += w1[t * C + j] * pooled[j];
    hid[t] = fmaxf(s, 0.f);
  }
  __syncthreads();
  if (t < C) {
    float s = b2[t];
    for (int j = 0; j < C4; ++j) s += w2[t * C4 + j] * hid[j];
    att[t] = 1.f / (1.f + expf(-s));
  }
}

__global__ void se_apply_kernel(const float* __restrict__ h,
                                const float* __restrict__ att,
                                const float* __restrict__ sc,
                                float* __restrict__ out, int HW, int total) {
  int i = blockIdx.x * 256 + threadIdx.x;
  if (i >= total) return;
  int c = i / HW;
  out[i] = dgelu(h[i] * att[c] + sc[i]);
}

__global__ void maxpool_kernel(const float* __restrict__ in,
                               float* __restrict__ out, int C, int H, int W) {
  int Ho = H >> 1, Wo = W >> 1;
  int total = C * Ho * Wo;
  int i = blockIdx.x * 256 + threadIdx.x;
  if (i >= total) return;
  int c = i / (Ho * Wo), rem = i - c * (Ho * Wo);
  int y = rem / Wo, x = rem - y * Wo;
  const float* p = in + (size_t)c * H * W + (size_t)(2 * y) * W + 2 * x;
  out[i] = fmaxf(fmaxf(p[0], p[1]), fmaxf(p[W], p[W + 1]));
}

__global__ void sort64_kernel(const float* __restrict__ sv,
                              float* __restrict__ out_s,
                              float* __restrict__ fs_tail) {
  __shared__ float a[64];
  int t = threadIdx.x;  // 64 threads
  a[t] = sv[t];
  __syncthreads();
  for (int k = 2; k <= 64; k <<= 1) {
    for (int j = k >> 1; j > 0; j >>= 1) {
      int ixj = t ^ j;
      if (ixj > t) {
        bool up = ((t & k) == 0);  // ascending network
        float x0 = a[t], x1 = a[ixj];
        if (up ? (x0 > x1) : (x0 < x1)) {
          a[t] = x1;
          a[ixj] = x0;
        }
      }
      __syncthreads();
    }
  }
  float v = fmaxf(a[63 - t], 1e-7f);  // descending + clamp
  out_s[t] = v;
  fs_tail[t] = v;
}

__global__ void copy_kernel(const float* __restrict__ src,
                            float* __restrict__ dst, int n) {
  int i = blockIdx.x * 256 + threadIdx.x;
  if (i < n) dst[i] = src[i];
}

// ---------------- complex 64x64 GEMM tile (one 16x16 tile per wave) --------
static __device__ inline void cgemm64(
    const float* __restrict__ Ar, const float* __restrict__ Ai,
    const float* __restrict__ Br, const float* __restrict__ Bi,
    float* __restrict__ Cr_, float* __restrict__ Ci_, int wv, int lane) {
  int m = lane & 15, kh = lane >> 4;
  int ti = wv >> 2, tj = wv & 3;
  v8f cr = {}, ci = {};
  for (int k0 = 0; k0 < 64; k0 += 4) {
    v2f ar, ai, br, bi, nbi;
    for (int j = 0; j < 2; ++j) {
      int k = k0 + 2 * kh + j;
      ar[j] = Ar[(ti * 16 + m) * 64 + k];
      ai[j] = Ai[(ti * 16 + m) * 64 + k];
      br[j] = Br[k * 64 + tj * 16 + m];
      bi[j] = Bi[k * 64 + tj * 16 + m];
    }
    nbi[0] = -bi[0];
    nbi[1] = -bi[1];
    cr = wmma4(ar, br, cr);   // + Ar*Br
    cr = wmma4(ai, nbi, cr);  // - Ai*Bi
    ci = wmma4(ar, bi, ci);   // + Ar*Bi
    ci = wmma4(ai, br, ci);   // + Ai*Br
  }
  for (int r = 0; r < 8; ++r) {
    int a = ti * 16 + kh * 8 + r, b = tj * 16 + m;
    Cr_[a * 64 + b] = cr[r];
    Ci_[a * 64 + b] = ci[r];
  }
}

// ---------------- persistent single-WG orthogonalization in LDS -----------
// Q (256x64 complex, interleaved re/im) DMA'd into LDS by the Tensor Data
// Mover; 10 outer polar iterations Q <- Q (Q^H Q)^(-1/2), inverse-sqrt via
// Newton-Schulz (same fixed point as the eigh formulation).
__global__ __launch_bounds__(512) void ortho_kernel(const float* __restrict__ Qin,
                                                    float* __restrict__ Qout) {
  extern __shared__ float smem_[];
  float* Qc = smem_;       // 32768 floats: Q interleaved (re,im) pairs
  float* Gr = Qc + 32768;  // 4096 each below
  float* Gi = Gr + 4096;
  float* Yr = Gi + 4096;
  float* Yi = Yr + 4096;
  float* Zr = Yi + 4096;
  float* Zi = Zr + 4096;
  float* Tr = Zi + 4096;
  float* Ti = Tr + 4096;
  float* Ur = Ti + 4096;
  float* Ui = Ur + 4096;
  float* scal = Ui + 4096;
  int tid = threadIdx.x, lane = tid & 31, wv = tid >> 5;  // 16 waves
  int m = lane & 15, kh = lane >> 4;

  // --- TDM: DMA the whole 128KB Q tile Global -> LDS (wave 0 issues) ---
  if (wv == 0) {
    unsigned long long ga = (unsigned long long)(__SIZE_TYPE__)Qin;
    u32x4 g0;
    g0[0] = 1u;  // count=1, user mode, no gather
    g0[1] = 0u;  // lds_addr = 0 (dynamic LDS base, no static LDS here)
    g0[2] = (unsigned int)(ga & 0xffffffffu);          // global_addr[31:0]
    g0[3] = (unsigned int)((ga >> 32) & 0x01ffffffu)   // global_addr[56:32]
            | (2u << 30);                              // type = 2 ("image")
    u32x8 g1;
    g1[0] = (2u << 16);                 // workgroup_mask=0, data_size=4B
    g1[1] = (32768u & 0xffffu) << 16;   // tensor_dim0[15:0] in bits[63:48]
    g1[2] = (32768u >> 16)              // tensor_dim0[31:16]
            | (1u << 16);               // tensor_dim1 = 1
    g1[3] = (32768u & 0xffffu) << 16;   // tile_dim0 = 32768 elements
    g1[4] = 1u;                         // tile_dim1 = 1, tile_dim2 = 0
    g1[5] = 32768u;                     // tensor_dim0_stride[31:0]
    g1[6] = 0u;                         // stride0 hi, stride1 lo
    g1[7] = 0u;                         // stride1 hi
    asm volatile("tensor_load_to_lds %0, %1" ::"s"(g0), "s"(g1) : "memory");
    __builtin_amdgcn_s_wait_tensorcnt(0);
  }
  __syncthreads();

  for (int it = 0; it < ORTHO_OUTER; ++it) {
    {  // G = Q^H Q  (K = 256), one 16x16 tile per wave
      int ti = wv >> 2, tj = wv & 3;
      v8f cr = {}, ci = {};
      for (int k0 = 0; k0 < 256; k0 += 4) {
        v2f ar, nai, ai, br, bi;
        for (int j = 0; j < 2; ++j) {
          int k = k0 + 2 * kh + j;
          int ia = 2 * (k * 64 + ti * 16 + m);
          int ib = 2 * (k * 64 + tj * 16 + m);
          ar[j] = Qc[ia];
          nai[j] = Qc[ia + 1];  // = -(conj imag)
          br[j] = Qc[ib];
          bi[j] = Qc[ib + 1];
        }
        ai[0] = -nai[0];
        ai[1] = -nai[1];
        cr = wmma4(ar, br, cr);
        cr = wmma4(nai, bi, cr);
        ci = wmma4(ar, bi, ci);
        ci = wmma4(ai, br, ci);
      }
      for (int r = 0; r < 8; ++r) {
        int a = ti * 16 + kh * 8 + r, b = tj * 16 + m;
        Gr[a * 64 + b] = cr[r];
        Gi[a * 64 + b] = ci[r];
      }
    }
    __syncthreads();
    if (tid == 0) {
      float t = 0.f;
      for (int a = 0; a < 64; ++a) t += Gr[a * 64 + a];
      scal[0] = fmaxf(t, 1e-12f);
    }
    __syncthreads();
    float invt = 1.0f / scal[0];
    for (int i = tid; i < 4096; i += 512) {  // Y = G/tr, Z = I
      Yr[i] = Gr[i] * invt;
      Yi[i] = Gi[i] * invt;
      int a = i >> 6, b = i & 63;
      Zr[i] = (a == b) ? 1.f : 0.f;
      Zi[i] = 0.f;
    }
    __syncthreads();
    for (int ns = 0; ns < ORTHO_NS; ++ns) {
      cgemm64(Zr, Zi, Yr, Yi, Tr, Ti, wv, lane);  // T = Z*Y
      __syncthreads();
      for (int i = tid; i < 4096; i += 512) {  // T = (3I - T)/2
        int a = i >> 6, b = i & 63;
        Tr[i] = 0.5f * (((a == b) ? 3.f : 0.f) - Tr[i]);
        Ti[i] = -0.5f * Ti[i];
      }
      __syncthreads();
      cgemm64(Yr, Yi, Tr, Ti, Ur, Ui, wv, lane);  // Ynew -> U
      cgemm64(Tr, Ti, Zr, Zi, Gr, Gi, wv, lane);  // Znew -> G (free)
      __syncthreads();
      for (int i = tid; i < 4096; i += 512) {
        Yr[i] = Ur[i];
        Yi[i] = Ui[i];
        Zr[i] = Gr[i];
        Zi[i] = Gi[i];
      }
      __syncthreads();
    }
    float isq = rsqrtf(scal[0]);  // G^(-1/2) = Z / sqrt(tr)
    {  // Q = (Q * Z) * isq ; wave wv owns row tile wv -> no cross-wave hazard
      v8f qcr[4], qci[4];
      for (int tjj = 0; tjj < 4; ++tjj) {
        v8f cr = {}, ci = {};
        for (int k0 = 0; k0 < 64; k0 += 4) {
          v2f ar, ai, br, bi, nbi;
          for (int j = 0; j < 2; ++j) {
            int k = k0 + 2 * kh + j;
            int ia = 2 * ((wv * 16 + m) * 64 + k);
            ar[j] = Qc[ia];
            ai[j] = Qc[ia + 1];
            br[j] = Zr[k * 64 + tjj * 16 + m];
            bi[j] = Zi[k * 64 + tjj * 16 + m];
          }
          nbi[0] = -bi[0];
          nbi[1] = -bi[1];
          cr = wmma4(ar, br, cr);
          cr = wmma4(ai, nbi, cr);
          ci = wmma4(ar, bi, ci);
          ci = wmma4(ai, br, ci);
        }
        qcr[tjj] = cr;
        qci[tjj] = ci;
      }
      for (int tjj = 0; tjj < 4; ++tjj)
        for (int r = 0; r < 8; ++r) {
          int row = wv * 16 + kh * 8 + r, col = tjj * 16 + m;
          Qc[2 * (row * 64 + col)] = qcr[tjj][r] * isq;
          Qc[2 * (row * 64 + col) + 1] = qci[tjj][r] * isq;
        }
    }
    __syncthreads();
  }
  for (int i = tid; i < 65536; i += 512) Qout[i] = Qc[i];
}

// ---------------- host orchestration ----------------
extern "C" void kernel_launch(void* const* d_in, const int* in_sizes, int n_in,
                              void* d_out, int out_size, void* d_ws,
                              size_t ws_size, hipStream_t stream) {
  (void)in_sizes; (void)n_in; (void)out_size; (void)ws_size;
  const float* X = (const float*)d_in[0];
  auto A = [&](int i) { return (const float*)d_in[i]; };
  float* ws = (float*)d_ws;
  const size_t BUF = 2097152;  // 32*256*256 floats
  float* B0 = ws;
  float* B1 = ws + BUF;
  float* B2 = ws + 2 * BUF;
  float* B3 = ws + 3 * BUF;
  float* SM = ws + 4 * BUF;
  float* pooled = SM;           // 256
  float* att = SM + 256;        // 256
  float* feat = SM + 512;       // 256
  float* qkv = SM + 768;        // 768
  float* feat2 = SM + 1536;     // 256
  float* s1 = SM + 1792;        // 512
  float* s2 = SM + 2304;        // 128
  float* sv = SM + 2432;        // 64
  float* fs = SM + 2496;        // 320
  float* uvec = SM + 2816;      // 1024
  float* vvec = SM + 3840;      // 1024
  float* Uraw = SM + 4864;      // 32768
  float* Vraw = SM + 37632;     // 32768
  float* outF = (float*)d_out;  // U(32768) | s(64) | V(32768)

  nhwc_to_chw_kernel<<<(65536 + 255) / 256, 256, 0, stream>>>(X, B3, 256, 256);

  auto run_block = [&](const float* in, int Cin, int Cout, int H, int W,
                       int base, float* h, float* h2, float* sc) {
    int HW = H * W;
    int nt = (HW / 16) * (Cout / 16);
    conv_wmma_kernel<<<(nt + 7) / 8, 256, 0, stream>>>(
        in, A(base + 0), A(base + 1), A(base + 2), A(base + 3), A(base + 4),
        A(base + 5), h, Cin, Cout, H, W, 3, 2);  // conv1+bn1+gelu
    conv_wmma_kernel<<<(nt + 7) / 8, 256, 0, stream>>>(
        h, A(base + 6), A(base + 7), A(base + 8), A(base + 9), A(base + 10),
        A(base + 11), h2, Cout, Cout, H, W, 3, 1);  // conv2+bn2
    conv_wmma_kernel<<<(nt + 7) / 8, 256, 0, stream>>>(
        in, A(base + 16), A(base + 17), nullptr, nullptr, nullptr, nullptr, sc,
        Cin, Cout, H, W, 1, 0);  // 1x1 residual
    channel_mean_kernel<<<Cout, 256, 0, stream>>>(h2, pooled, HW);
    se_fc_kernel<<<1, 256, 0, stream>>>(pooled, A(base + 12), A(base + 13),
                                        A(base + 14), A(base + 15), att, Cout,
                                        Cout / 4);
    int total = Cout * HW;
    se_apply_kernel<<<(total + 255) / 256, 256, 0, stream>>>(h2, att, sc, sc,
                                                             HW, total);
  };

  run_block(B3, 2, 32, 256, 256, 1, B0, B1, B2);
  maxpool_kernel<<<(32 * 128 * 128 + 255) / 256, 256, 0, stream>>>(B2, B0, 32, 256, 256);
  run_block(B0, 32, 64, 128, 128, 19, B1, B2, B3);
  maxpool_kernel<<<(64 * 64 * 64 + 255) / 256, 256, 0, stream>>>(B3, B1, 64, 128, 128);
  run_block(B1, 64, 128, 64, 64, 37, B2, B3, B0);
  maxpool_kernel<<<(128 * 32 * 32 + 255) / 256, 256, 0, stream>>>(B0, B2, 128, 64, 64);
  run_block(B2, 128, 256, 32, 32, 55, B3, B0, B1);
  channel_mean_kernel<<<256, 256, 0, stream>>>(B1, feat, 1024);

  // attention head: softmax over singleton axis == 1  =>  ctx = v = qkv[2E:3E]
  gemv_kernel<<<768 / 8, 256, 0, stream>>>(A(73), A(74), feat, qkv, 256, 768, 0);
  gemv_kernel<<<256 / 8, 256, 0, stream>>>(A(75), A(76), qkv + 512, feat2, 256, 256, 0);
  // singular-value path
  gemv_kernel<<<512 / 8, 256, 0, stream>>>(A(77), A(78), feat2, s1, 256, 512, 1);
  gemv_kernel<<<128 / 8, 256, 0, stream>>>(A(79), A(80), s1, s2, 512, 128, 1);
  gemv_kernel<<<64 / 8, 256, 0, stream>>>(A(81), A(82), s2, sv, 128, 64, 3);
  sort64_kernel<<<1, 64, 0, stream>>>(sv, outF + 32768, fs + 256);
  copy_kernel<<<1, 256, 0, stream>>>(feat2, fs, 256);
  // U path (u2_w is 134 MB -> HBM-bound streamer)
  gemv_kernel<<<1024 / 8, 256, 0, stream>>>(A(83), A(84), fs, uvec, 320, 1024, 1);
  gemv_kernel<<<32768 / 8, 256, 0, stream>>>(A(85), A(86), uvec, Uraw, 1024, 32768, 0);
  // V path
  gemv_kernel<<<1024 / 8, 256, 0, stream>>>(A(87), A(88), fs, vvec, 320, 1024, 1);
  gemv_kernel<<<32768 / 8, 256, 0, stream>>>(A(89), A(90), vvec, Vraw, 1024, 32768, 0);
  // orthogonalize (single WGP, Q + work matrices resident in 320KB LDS)
  size_t smem = (size_t)(32768 + 10 * 4096 + 16) * sizeof(float);
  ortho_kernel<<<1, 512, smem, stream>>>(Uraw, outF);
  ortho_kernel<<<1, 512, smem, stream>>>(Vraw, outF + 32832);
}